// RealtimeNgramProcessor_47090021434018
// MI455X (gfx1250) — compile-verified
//
#include <hip/hip_runtime.h>
#include <stdint.h>

// Problem constants from the reference
#define BASEV 50257ull
#define BDIM  512
#define SDIM  8192
#define TILE  256
#define NB    (1u << 18)   // buckets per table; 3*(NB+1)*4B = ~3.15 MB in d_ws

static_assert(SDIM % TILE == 0, "tile must divide S");

__host__ __device__ constexpr unsigned long long kpow(int n) {
  unsigned long long r = 1;
  for (int i = 0; i < n; ++i) r *= BASEV;
  return r;                       // 50257^4 = 6.38e18 < 2^63, fits u64
}
__host__ __device__ constexpr unsigned long long dstep(int n) {
  return (kpow(n) + NB - 1) / NB; // compile-time constant -> magic-number division
}

// ---------------------------------------------------------------------------
// Pass 1: bucket index. start[b] = lower_bound(keys, b*D). start[NB] == V.
// 786k threads total, each an L2-resident binary search -> negligible cost.
// ---------------------------------------------------------------------------
__global__ void build_bucket_starts(const long long* __restrict__ keys, int V,
                                    unsigned* __restrict__ start,
                                    unsigned long long D) {
  unsigned b = blockIdx.x * blockDim.x + threadIdx.x;
  if (b > NB) return;
  unsigned long long target = (unsigned long long)b * D;
  int lo = 0, hi = V;
  while (lo < hi) {
    int mid = (lo + hi) >> 1;
    if ((unsigned long long)keys[mid] < target) lo = mid + 1;
    else hi = mid;
  }
  start[b] = (unsigned)lo;
}

// Scan one bucket (avg ~1.5 keys). keys sorted -> break on k >= w.
__device__ __forceinline__ long long scan_bucket(unsigned long long w,
                                                 unsigned s, unsigned e,
                                                 const long long* __restrict__ keys,
                                                 const long long* __restrict__ ids) {
  long long r = 0;
  for (unsigned i = s; i < e; ++i) {
    unsigned long long k = (unsigned long long)keys[i];
    if (k >= w) { if (k == w) r = ids[i]; break; }
  }
  return r;
}

// Reference-equivalent binary search (fallback path, no workspace needed).
__device__ __forceinline__ long long bs_lookup(unsigned long long w,
                                               const long long* __restrict__ keys,
                                               const long long* __restrict__ ids, int V) {
  int lo = 0, hi = V;
  while (lo < hi) {
    int mid = (lo + hi) >> 1;
    if ((unsigned long long)keys[mid] < w) lo = mid + 1;
    else hi = mid;
  }
  return (lo < V && (unsigned long long)keys[lo] == w) ? ids[lo] : 0ll;
}

// ---------------------------------------------------------------------------
// Pass 2: main lookup. One thread per position; x tile staged via LDS so the
// 4 overlapping tokens per lane come from ds_load_b64, not 4x VMEM traffic.
// ---------------------------------------------------------------------------
__global__ void __launch_bounds__(TILE)
ngram_lookup_buckets(const long long* __restrict__ x,
                     const long long* __restrict__ k2, const long long* __restrict__ i2,
                     const long long* __restrict__ k3, const long long* __restrict__ i3,
                     const long long* __restrict__ k4, const long long* __restrict__ i4,
                     const unsigned* __restrict__ s2t, const unsigned* __restrict__ s3t,
                     const unsigned* __restrict__ s4t,
                     long long* __restrict__ out) {
  __shared__ long long tile[TILE + 3];
  const int row = blockIdx.y;
  const int t0  = blockIdx.x * TILE;
  const int tid = threadIdx.x;
  const long long* xrow = x + (size_t)row * SDIM;

  tile[3 + tid] = xrow[t0 + tid];
  if (tid < 3) {                      // left halo; rows are zero-padded on the left
    const int src = t0 + tid - 3;
    tile[tid] = (src >= 0) ? xrow[src] : 0ll;
  }
  if (t0 + TILE < SDIM)               // gfx1250 global_prefetch_b8 for next tile
    __builtin_prefetch(xrow + t0 + TILE + tid, 0, 1);
  __syncthreads();

  const unsigned long long a0 = (unsigned long long)tile[3 + tid]; // x[t]
  const unsigned long long a1 = (unsigned long long)tile[2 + tid]; // x[t-1]
  const unsigned long long a2 = (unsigned long long)tile[1 + tid]; // x[t-2]
  const unsigned long long a3 = (unsigned long long)tile[0 + tid]; // x[t-3]

  const unsigned long long w2 = a1 * BASEV + a0;
  const unsigned long long w3 = a2 * (BASEV * BASEV) + w2;
  const unsigned long long w4 = a3 * (BASEV * BASEV * BASEV) + w3;

  // Buckets via constant division (mulhi magic sequence, no runtime divide).
  unsigned b2 = (unsigned)(w2 / dstep(2)); if (b2 >= NB) b2 = NB - 1;
  unsigned b3 = (unsigned)(w3 / dstep(3)); if (b3 >= NB) b3 = NB - 1;
  unsigned b4 = (unsigned)(w4 / dstep(4)); if (b4 >= NB) b4 = NB - 1;

  // Issue all six start loads up front so the three chains overlap in flight.
  const unsigned lo2 = s2t[b2], hi2 = s2t[b2 + 1];
  const unsigned lo3 = s3t[b3], hi3 = s3t[b3 + 1];
  const unsigned lo4 = s4t[b4], hi4 = s4t[b4 + 1];

  const long long r2 = scan_bucket(w2, lo2, hi2, k2, i2);
  const long long r3 = scan_bucket(w3, lo3, hi3, k3, i3);
  const long long r4 = scan_bucket(w4, lo4, hi4, k4, i4);

  const size_t pos   = (size_t)row * SDIM + (size_t)(t0 + tid);
  const size_t plane = (size_t)BDIM * SDIM;
  out[pos]             = r2;
  out[plane + pos]     = r3;
  out[2 * plane + pos] = r4;
}

// Fallback: same tiling, plain binary search (used only if ws too small).
__global__ void __launch_bounds__(TILE)
ngram_lookup_bsearch(const long long* __restrict__ x,
                     const long long* __restrict__ k2, const long long* __restrict__ i2,
                     const long long* __restrict__ k3, const long long* __restrict__ i3,
                     const long long* __restrict__ k4, const long long* __restrict__ i4,
                     int V2, int V3, int V4, long long* __restrict__ out) {
  __shared__ long long tile[TILE + 3];
  const int row = blockIdx.y;
  const int t0  = blockIdx.x * TILE;
  const int tid = threadIdx.x;
  const long long* xrow = x + (size_t)row * SDIM;

  tile[3 + tid] = xrow[t0 + tid];
  if (tid < 3) {
    const int src = t0 + tid - 3;
    tile[tid] = (src >= 0) ? xrow[src] : 0ll;
  }
  __syncthreads();

  const unsigned long long a0 = (unsigned long long)tile[3 + tid];
  const unsigned long long a1 = (unsigned long long)tile[2 + tid];
  const unsigned long long a2 = (unsigned long long)tile[1 + tid];
  const unsigned long long a3 = (unsigned long long)tile[0 + tid];
  const unsigned long long w2 = a1 * BASEV + a0;
  const unsigned long long w3 = a2 * (BASEV * BASEV) + w2;
  const unsigned long long w4 = a3 * (BASEV * BASEV * BASEV) + w3;

  const size_t pos   = (size_t)row * SDIM + (size_t)(t0 + tid);
  const size_t plane = (size_t)BDIM * SDIM;
  out[pos]             = bs_lookup(w2, k2, i2, V2);
  out[plane + pos]     = bs_lookup(w3, k3, i3, V3);
  out[2 * plane + pos] = bs_lookup(w4, k4, i4, V4);
}

extern "C" void kernel_launch(void* const* d_in, const int* in_sizes, int n_in,
                              void* d_out, int out_size, void* d_ws, size_t ws_size,
                              hipStream_t stream) {
  const long long* x  = (const long long*)d_in[0];
  const long long* k2 = (const long long*)d_in[1];
  const long long* i2 = (const long long*)d_in[2];
  const long long* k3 = (const long long*)d_in[3];
  const long long* i3 = (const long long*)d_in[4];
  const long long* k4 = (const long long*)d_in[5];
  const long long* i4 = (const long long*)d_in[6];
  const int V2 = in_sizes[1], V3 = in_sizes[3], V4 = in_sizes[5];
  long long* out = (long long*)d_out;

  dim3 grid(SDIM / TILE, BDIM);
  const size_t per  = (size_t)(NB + 1) * sizeof(unsigned);
  const size_t need = 3 * per;

  if (ws_size >= need) {
    unsigned* s2t = (unsigned*)d_ws;
    unsigned* s3t = s2t + (NB + 1);
    unsigned* s4t = s3t + (NB + 1);
    const int bb = (int)((NB + 1 + 255) / 256);
    build_bucket_starts<<<bb, 256, 0, stream>>>(k2, V2, s2t, dstep(2));
    build_bucket_starts<<<bb, 256, 0, stream>>>(k3, V3, s3t, dstep(3));
    build_bucket_starts<<<bb, 256, 0, stream>>>(k4, V4, s4t, dstep(4));
    ngram_lookup_buckets<<<grid, TILE, 0, stream>>>(x, k2, i2, k3, i3, k4, i4,
                                                    s2t, s3t, s4t, out);
  } else {
    ngram_lookup_bsearch<<<grid, TILE, 0, stream>>>(x, k2, i2, k3, i3, k4, i4,
                                                    V2, V3, V4, out);
  }
}